// DropToken_22428319220330
// MI455X (gfx1250) — compile-verified
//
#include <hip/hip_runtime.h>

typedef unsigned int u32;
typedef unsigned long long u64;
typedef u32 v4u __attribute__((ext_vector_type(4)));
typedef int v4i __attribute__((ext_vector_type(4)));
typedef int v8i __attribute__((ext_vector_type(8)));

namespace {
constexpr int kBatch     = 4;
constexpr int kLength    = 8192;
constexpr int kEmbed     = 1024;
constexpr int kCap       = 4096;
constexpr int kRows      = 16;                 // rows per TDM gather descriptor (16-bit idx mode max)
constexpr int kChunks    = kCap / kRows;       // 256 chunks per batch
constexpr int kTileElems = kRows * kEmbed;     // 16384 f32 = 64 KB LDS tile
}

__global__ __launch_bounds__(32)
void drop_token_tdm_gather(const float* __restrict__ in,
                           const int*  __restrict__ idx,
                           float* __restrict__ out)
{
  __shared__ float tile[kTileElems];           // 64 KB staging tile

  const u32 wg = blockIdx.x;                   // 0..1023
  const u32 b  = wg / kChunks;
  const u32 c0 = (wg % kChunks) * kRows;

  // Pack 16 gather row indices (all < 8192, fit 16 bits), 2 per dword,
  // in D# group2/group3 layout for 16-bit index gather mode.
  u32 packed[8];
#pragma unroll
  for (int i = 0; i < 8; ++i) {
    u32 lo = (u32)idx[c0 + 2 * i]     & 0xFFFFu;
    u32 hi = (u32)idx[c0 + 2 * i + 1] & 0xFFFFu;
    packed[i] = lo | (hi << 16);
  }

  // Low 32 bits of the flat shared-memory address == LDS byte address.
  const u32 lds_addr = (u32)(u64)(void*)tile;

  const v8i gz8 = {0, 0, 0, 0, 0, 0, 0, 0};    // unused VADDR4 slot
  const v4i gz4 = {0, 0, 0, 0};

  // ---------------- TDM gather load: 16 indexed rows -> LDS ----------------
  const u64 ga = (u64)(const void*)(in + (u64)b * kLength * kEmbed);
  v4u g0l;
  g0l[0] = 0x80000001u;                            // count=1 | gather_mode=1 | 16-bit idx
  g0l[1] = lds_addr;                               // lds_addr [63:32]
  g0l[2] = (u32)(ga & 0xFFFFFFFFu);                // global_addr [95:64]
  g0l[3] = (u32)((ga >> 32) & 0x01FFFFFFu)         // global_addr [120:96]
         | (2u << 30);                             // type=2 ("image")

  v8i g1l;
  g1l[0] = (int)(2u << 16);                        // wg_mask=0, data_size=2 (4 bytes)
  g1l[1] = (int)((u32)(kEmbed & 0xFFFF) << 16);    // tensor_dim0[15:0] = 1024
  g1l[2] = (int)(((u32)kEmbed >> 16)               // tensor_dim0[31:16]
         | ((u32)(kLength & 0xFFFF) << 16));       // tensor_dim1[15:0] = 8192
  g1l[3] = (int)(((u32)kLength >> 16)              // tensor_dim1[31:16]
         | ((u32)(kEmbed & 0xFFFF) << 16));        // tile_dim0 = 1024 (full row)
  g1l[4] = kRows;                                  // tile_dim1 = #valid indices = 16
  g1l[5] = kEmbed;                                 // tensor_dim0_stride[31:0] = 1024
  g1l[6] = 0;                                      // stride[47:32], dim1_stride (ignored)
  g1l[7] = 0;

  v4i g2l = { (int)packed[0], (int)packed[1], (int)packed[2], (int)packed[3] };
  v4i g3l = { (int)packed[4], (int)packed[5], (int)packed[6], (int)packed[7] };

  __builtin_amdgcn_tensor_load_to_lds(g0l, g1l, g2l, g3l, gz8, 0);
  __builtin_amdgcn_s_wait_tensorcnt(0);            // tile resident in LDS

  // ------------- TDM store: LDS tile -> contiguous out rows ---------------
  const u64 go = (u64)(void*)(out + ((u64)b * kCap + c0) * kEmbed);
  v4u g0s;
  g0s[0] = 1u;                                     // count=1, normal (non-gather) mode
  g0s[1] = lds_addr;
  g0s[2] = (u32)(go & 0xFFFFFFFFu);
  g0s[3] = (u32)((go >> 32) & 0x01FFFFFFu) | (2u << 30);

  v8i g1s;                                         // 1-D tile of 16384 f32 (64 KB)
  g1s[0] = (int)(2u << 16);                        // data_size = 4 bytes
  g1s[1] = (int)((u32)(kTileElems & 0xFFFF) << 16);// tensor_dim0[15:0] = 16384
  g1s[2] = (int)(((u32)kTileElems >> 16)           // tensor_dim0[31:16]
         | (1u << 16));                            // tensor_dim1 = 1
  g1s[3] = (int)((u32)(kTileElems & 0xFFFF) << 16);// tile_dim0 = 16384
  g1s[4] = 0;                                      // tile_dim1/2 unused (1-D)
  g1s[5] = kTileElems;                             // tensor_dim0_stride (unused for 1-D)
  g1s[6] = 0;
  g1s[7] = 0;

  __builtin_amdgcn_tensor_store_from_lds(g0s, g1s, gz4, gz4, gz8, 0);
  // S_ENDPGM performs an implicit wait-idle (covers outstanding TENSORcnt).
}

extern "C" void kernel_launch(void* const* d_in, const int* in_sizes, int n_in,
                              void* d_out, int out_size, void* d_ws, size_t ws_size,
                              hipStream_t stream) {
  const float* in  = (const float*)d_in[0];   // [4, 8192, 1024] f32
  const int*   idx = (const int*)d_in[1];     // [4096] i32
  float* out = (float*)d_out;                 // [4, 4096, 1024] f32

  dim3 grid(kBatch * kChunks);                // 1024 workgroups
  dim3 block(32);                             // one wave32 each drives the TDM
  drop_token_tdm_gather<<<grid, block, 0, stream>>>(in, idx, out);
}